// GCNNet_88630945120525
// MI455X (gfx1250) — compile-verified
//
#include <hip/hip_runtime.h>
#include <hip/hip_bf16.h>
#include <math.h>

// ---------------------------------------------------------------------------
// GAT / GCN-attention network for gfx1250 (MI455X).
//  - Dense GEMMs: V_WMMA_F32_16X16X4_F32, wave32. W staged in LDS (<=64KB of
//    the 320KB WGP LDS); each wave computes a 16x64 stripe (4 WMMA per A-load).
//  - Edge softmax + aggregation via L2-resident f32 atomics (z fits in 192MB L2).
// ---------------------------------------------------------------------------

#define N_NODES 100000
#define N_EDGES 1600000
#define IN_DIM  64
#define HID     64
#define HEADS   4
#define OUT_DIM 7

typedef __attribute__((ext_vector_type(2))) float v2f;
typedef __attribute__((ext_vector_type(8))) float v8f;

// ---------------- WMMA fp32 GEMM (dout fixed = 64):
//   Z[:, colz:colz+64] = X(N,K) * W(K,64) + b
// Block = 128 threads (4 waves). Block covers a 64-row stripe; wave wy owns
// rows [row0, row0+16) x all 4 column tiles. W staged in LDS (K*64 floats).
// Requires N%16==0 (100000 ok), K%4==0.
__global__ __launch_bounds__(128)
void gemm_wmma_f32_d64(const float* __restrict__ X, int ldx,
                       const float* __restrict__ W,      // [K, 64] row-major
                       const float* __restrict__ bias,   // [64]
                       float* __restrict__ Z, int ldz, int colz,
                       int N, int K)
{
    extern __shared__ float lds[];                 // K*64 floats
    const int tid  = threadIdx.x;
    const int lane = tid & 31;
    const int wy   = tid >> 5;                     // wave id 0..3 -> row tile
    const int m    = lane & 15;
    const int kh   = lane >> 4;                    // lane half: K offset 0 / +2

    // ---- cooperative stage of W into LDS (contiguous K*64 floats)
    {
        const int total4 = (K * 64) >> 2;          // float4 count
        const float4* __restrict__ w4 = (const float4*)W;
        float4* l4 = (float4*)lds;
        for (int i = tid; i < total4; i += 128) l4[i] = w4[i];
    }
    __syncthreads();

    const int row0 = blockIdx.x * 64 + wy * 16;
    if (row0 >= N) return;

    const float* __restrict__ xrow = X + (size_t)(row0 + m) * ldx;

    v8f acc[4] = {v8f{}, v8f{}, v8f{}, v8f{}};
    for (int k = 0; k < K; k += 4) {
        // A fragment (16x4 f32): lanes 0-15 K=k..k+1, lanes 16-31 K=k+2..k+3
        v2f a;
        a.x = xrow[k + 2 * kh + 0];
        a.y = xrow[k + 2 * kh + 1];
        if (k + 8 < K) __builtin_prefetch(&xrow[k + 8], 0, 0);  // global_prefetch_b8
        // 4 column tiles reuse the same A fragment; B fragments from LDS
        #pragma unroll
        for (int ct = 0; ct < 4; ++ct) {
            v2f b;
            b.x = lds[(k + 2 * kh + 0) * 64 + ct * 16 + m];
            b.y = lds[(k + 2 * kh + 1) * 64 + ct * 16 + m];
            acc[ct] = __builtin_amdgcn_wmma_f32_16x16x4_f32(
                          false, a, false, b, (short)0, acc[ct], false, false);
        }
    }

    // C/D layout: VGPR r -> M=row0+r (lanes 0-15) / row0+r+8 (lanes 16-31), N=m
    #pragma unroll
    for (int ct = 0; ct < 4; ++ct) {
        const float bj = bias[ct * 16 + m];
        #pragma unroll
        for (int r = 0; r < 8; ++r) {
            int row = row0 + r + 8 * kh;
            Z[(size_t)row * ldz + colz + ct * 16 + m] = acc[ct][r] + bj;
        }
    }
}

// ---------------- small GEMM for the 64->7 output heads (thread per (n,j))
__global__ void gemm_small(const float* __restrict__ X, int ldx,
                           const float* __restrict__ W,   // [K, dout]
                           const float* __restrict__ bias,
                           float* __restrict__ Z, int ldz,
                           int N, int K, int dout)
{
    int t = blockIdx.x * blockDim.x + threadIdx.x;
    if (t >= N * dout) return;
    int n = t / dout, j = t - n * dout;
    float acc = bias[j];
    const float* __restrict__ xr = X + (size_t)n * ldx;
    for (int k = 0; k < K; ++k) acc = fmaf(xr[k], W[(size_t)k * dout + j], acc);
    Z[(size_t)n * ldz + j] = acc;
}

// ---------------- per-node attention logits: sl=z.a[:d], sr=z.a[d:]
__global__ void attn_scores(const float* __restrict__ Z, int ldz,
                            const float* __restrict__ a, int d,
                            float* __restrict__ sl, float* __restrict__ sr, int N)
{
    int n = blockIdx.x * blockDim.x + threadIdx.x;
    if (n >= N) return;
    const float* __restrict__ zr = Z + (size_t)n * ldz;
    float l = 0.f, r = 0.f;
    if (((d | ldz) & 3) == 0) {
        for (int j = 0; j < d; j += 4) {
            float4 z4 = *(const float4*)&zr[j];          // global_load_b128
            float4 al = *(const float4*)&a[j];
            float4 ar = *(const float4*)&a[d + j];
            l = fmaf(z4.x, al.x, fmaf(z4.y, al.y, fmaf(z4.z, al.z, fmaf(z4.w, al.w, l))));
            r = fmaf(z4.x, ar.x, fmaf(z4.y, ar.y, fmaf(z4.z, ar.z, fmaf(z4.w, ar.w, r))));
        }
    } else {
        for (int j = 0; j < d; ++j) {
            float z = zr[j];
            l = fmaf(z, a[j], l);
            r = fmaf(z, a[d + j], r);
        }
    }
    sl[n] = l;
    sr[n] = r;
}

// ---------------- fill helper
__global__ void fill_f32(float* __restrict__ p, float v, long long n)
{
    long long i = (long long)blockIdx.x * blockDim.x + threadIdx.x;
    if (i < n) p[i] = v;
}

// ---------------- edge scores: e = GELU_exact(sl[src]+sr[dst]); atomic max per dst
__global__ void edge_gelu_max(const float* __restrict__ sl, const float* __restrict__ sr,
                              const int* __restrict__ src, const int* __restrict__ dst,
                              float* __restrict__ eb, float* __restrict__ mx, int E)
{
    int e = blockIdx.x * blockDim.x + threadIdx.x;
    if (e >= E) return;
    float x = sl[src[e]] + sr[dst[e]];
    float g = 0.5f * x * (1.0f + erff(x * 0.70710678118654752f));
    eb[e] = g;
    int d = dst[e];
    // float atomic max via signed/unsigned integer ordering trick
    if (g >= 0.0f) atomicMax((int*)&mx[d], __float_as_int(g));
    else           atomicMin((unsigned int*)&mx[d], __float_as_uint(g));
}

// m = isfinite(m) ? m : 0   (matches reference's where(isfinite(m), m, 0))
__global__ void fix_max(float* __restrict__ mx, int N)
{
    int n = blockIdx.x * blockDim.x + threadIdx.x;
    if (n >= N) return;
    float v = mx[n];
    if (!isfinite(v)) mx[n] = 0.0f;
}

// ex = exp(e - m[dst]); denom[dst] += ex
__global__ void edge_exp_sum(float* __restrict__ eb, const float* __restrict__ mx,
                             const int* __restrict__ dst, float* __restrict__ den, int E)
{
    int e = blockIdx.x * blockDim.x + threadIdx.x;
    if (e >= E) return;
    int d = dst[e];
    float ex = __expf(eb[e] - mx[d]);
    eb[e] = ex;
    atomicAdd(&den[d], ex);
}

// ---- d=64 aggregation: thread per (edge, quad-of-4); b128 gather + 4 atomics
// H[dst, colh + 4q..4q+3] += scale * (ex/denom[dst]) * Z[src, 4q..4q+3]
__global__ void aggregate64(const float* __restrict__ eb, const float* __restrict__ den,
                            const int* __restrict__ src, const int* __restrict__ dst,
                            const float* __restrict__ Z, int ldz,
                            float* __restrict__ H, int ldh, int colh,
                            float scale, long long total)   // total = E*16
{
    long long t = (long long)blockIdx.x * blockDim.x + threadIdx.x;
    if (t >= total) return;
    int e = (int)(t >> 4);
    int q = (int)(t & 15);
    int s = src[e], dd = dst[e];
    float w = scale * eb[e] / den[dd];
    float4 z4 = *(const float4*)&Z[(size_t)s * ldz + 4 * q];   // L2-resident gather
    float* hp = &H[(size_t)dd * ldh + colh + 4 * q];
    atomicAdd(hp + 0, w * z4.x);
    atomicAdd(hp + 1, w * z4.y);
    atomicAdd(hp + 2, w * z4.z);
    atomicAdd(hp + 3, w * z4.w);
}

// ---- generic scalar aggregation (used for d=7 out heads)
__global__ void aggregate(const float* __restrict__ eb, const float* __restrict__ den,
                          const int* __restrict__ src, const int* __restrict__ dst,
                          const float* __restrict__ Z, int ldz,
                          float* __restrict__ H, int ldh, int colh,
                          int d, int logdp, float scale, long long total)
{
    long long t = (long long)blockIdx.x * blockDim.x + threadIdx.x;
    if (t >= total) return;
    int e = (int)(t >> logdp);
    int j = (int)(t & ((1 << logdp) - 1));
    if (j >= d) return;
    int s = src[e], dd = dst[e];
    float alpha = eb[e] / den[dd];
    atomicAdd(&H[(size_t)dd * ldh + colh + j], scale * alpha * Z[(size_t)s * ldz + j]);
}

// ---------------- final 7x7 linear + row softmax
__global__ void final_linear_softmax(const float* __restrict__ outs,  // [N,7]
                                     const float* __restrict__ linW,  // [7,7] (in,out)
                                     const float* __restrict__ linb,
                                     float* __restrict__ out, int N)
{
    int n = blockIdx.x * blockDim.x + threadIdx.x;
    if (n >= N) return;
    float v[OUT_DIM];
    #pragma unroll
    for (int j = 0; j < OUT_DIM; ++j) v[j] = linb[j];
    #pragma unroll
    for (int i = 0; i < OUT_DIM; ++i) {
        float x = outs[(size_t)n * OUT_DIM + i];
        #pragma unroll
        for (int j = 0; j < OUT_DIM; ++j) v[j] = fmaf(x, linW[i * OUT_DIM + j], v[j]);
    }
    float m = v[0];
    #pragma unroll
    for (int j = 1; j < OUT_DIM; ++j) m = fmaxf(m, v[j]);
    float s = 0.f;
    #pragma unroll
    for (int j = 0; j < OUT_DIM; ++j) { v[j] = __expf(v[j] - m); s += v[j]; }
    float inv = 1.0f / s;
    #pragma unroll
    for (int j = 0; j < OUT_DIM; ++j) out[(size_t)n * OUT_DIM + j] = v[j] * inv;
}

// ---------------------------------------------------------------------------
// Host-side orchestration
// ---------------------------------------------------------------------------
namespace {

struct Ws {
    float* h;     // [N, 256]   h0 / h1 (shared)
    float* f;     // [N, 64]    f1 / f2 (shared)
    float* z;     // [N, 64]    per-head z (also used with ld=7 for out heads)
    float* sl;    // [N]
    float* sr;    // [N]
    float* mx;    // [N]
    float* den;   // [N]
    float* eb;    // [E]
    float* outs;  // [N, 7]
};

inline Ws carve(void* ws) {
    float* p = (float*)ws;
    Ws w;
    w.h    = p; p += (size_t)N_NODES * 256;
    w.f    = p; p += (size_t)N_NODES * 64;
    w.z    = p; p += (size_t)N_NODES * 64;
    w.sl   = p; p += N_NODES;
    w.sr   = p; p += N_NODES;
    w.mx   = p; p += N_NODES;
    w.den  = p; p += N_NODES;
    w.eb   = p; p += N_EDGES;
    w.outs = p; p += (size_t)N_NODES * OUT_DIM;
    return w;
}

// Shared softmax/aggregate pipeline over edges. d==64 uses the float4 path.
inline void edge_softmax_aggregate(const Ws& w, const int* src, const int* dst,
                                   const float* a_vec, int d, int ldz,
                                   float* H, int ldh, int colh, float scale,
                                   hipStream_t stream)
{
    const int NB = (N_NODES + 255) / 256;
    const int EB = (N_EDGES + 255) / 256;
    attn_scores<<<NB, 256, 0, stream>>>(w.z, ldz, a_vec, d, w.sl, w.sr, N_NODES);
    fill_f32<<<NB, 256, 0, stream>>>(w.mx, -INFINITY, N_NODES);
    fill_f32<<<NB, 256, 0, stream>>>(w.den, 0.0f, N_NODES);
    edge_gelu_max<<<EB, 256, 0, stream>>>(w.sl, w.sr, src, dst, w.eb, w.mx, N_EDGES);
    fix_max<<<NB, 256, 0, stream>>>(w.mx, N_NODES);
    edge_exp_sum<<<EB, 256, 0, stream>>>(w.eb, w.mx, dst, w.den, N_EDGES);
    if (d == 64) {
        long long total = (long long)N_EDGES * 16;
        int AB = (int)((total + 255) / 256);
        aggregate64<<<AB, 256, 0, stream>>>(w.eb, w.den, src, dst, w.z, ldz,
                                            H, ldh, colh, scale, total);
    } else {
        long long total = (long long)N_EDGES * 8;       // dpad=8 for d=7
        int AB = (int)((total + 255) / 256);
        aggregate<<<AB, 256, 0, stream>>>(w.eb, w.den, src, dst, w.z, ldz,
                                          H, ldh, colh, d, 3, scale, total);
    }
}

} // namespace

extern "C" void kernel_launch(void* const* d_in, const int* in_sizes, int n_in,
                              void* d_out, int out_size, void* d_ws, size_t ws_size,
                              hipStream_t stream)
{
    (void)in_sizes; (void)n_in; (void)out_size; (void)ws_size;

    const float* feature = (const float*)d_in[0];
    const int*   src     = (const int*)  d_in[1];
    const int*   dst     = (const int*)  d_in[2];
    const float* W0  = (const float*)d_in[3];   // [4,64,64]
    const float* b0  = (const float*)d_in[4];   // [4,64]
    const float* a0  = (const float*)d_in[5];   // [4,128]
    const float* W0h = (const float*)d_in[6];   // [256,64]
    const float* b0h = (const float*)d_in[7];   // [64]
    const float* a0h = (const float*)d_in[8];   // [128]
    const float* W1  = (const float*)d_in[9];   // [4,64,64]
    const float* b1  = (const float*)d_in[10];  // [4,64]
    const float* a1  = (const float*)d_in[11];  // [4,128]
    const float* W1h = (const float*)d_in[12];  // [256,64]
    const float* b1h = (const float*)d_in[13];  // [64]
    const float* a1h = (const float*)d_in[14];  // [128]
    const float* Wo  = (const float*)d_in[15];  // [4,64,7]
    const float* bo  = (const float*)d_in[16];  // [4,7]
    const float* ao  = (const float*)d_in[17];  // [4,14]
    const float* linW = (const float*)d_in[18]; // [7,7]
    const float* linb = (const float*)d_in[19]; // [7]
    float* out = (float*)d_out;

    Ws w = carve(d_ws);

    const int NB = (N_NODES + 255) / 256;
    const int gemmBlocks = (N_NODES + 63) / 64;          // 64-row stripes
    const size_t lds64  = (size_t)64  * 64 * sizeof(float);   // 16 KB (K=64)
    const size_t lds256 = (size_t)256 * 64 * sizeof(float);   // 64 KB (K=256)

    // ---------------- layer 0: 4 heads IN(64)->HID(64), concat into h0[N,256]
    fill_f32<<<(int)(((long long)N_NODES * 256 + 255) / 256), 256, 0, stream>>>(
        w.h, 0.0f, (long long)N_NODES * 256);
    for (int i = 0; i < HEADS; ++i) {
        gemm_wmma_f32_d64<<<gemmBlocks, 128, lds64, stream>>>(
            feature, IN_DIM, W0 + (size_t)i * IN_DIM * HID, b0 + i * HID,
            w.z, HID, 0, N_NODES, IN_DIM);
        edge_softmax_aggregate(w, src, dst, a0 + i * 2 * HID, HID, HID,
                               w.h, 256, i * HID, 1.0f, stream);
    }

    // ---------------- f1 = gcn_attn(h0[N,256], W0h[256,64])
    fill_f32<<<(int)(((long long)N_NODES * 64 + 255) / 256), 256, 0, stream>>>(
        w.f, 0.0f, (long long)N_NODES * 64);
    gemm_wmma_f32_d64<<<gemmBlocks, 128, lds256, stream>>>(
        w.h, 256, W0h, b0h, w.z, HID, 0, N_NODES, 4 * HID);
    edge_softmax_aggregate(w, src, dst, a0h, HID, HID, w.f, 64, 0, 1.0f, stream);

    // ---------------- layer 1: 4 heads HID->HID, concat into h1[N,256]
    fill_f32<<<(int)(((long long)N_NODES * 256 + 255) / 256), 256, 0, stream>>>(
        w.h, 0.0f, (long long)N_NODES * 256);
    for (int i = 0; i < HEADS; ++i) {
        gemm_wmma_f32_d64<<<gemmBlocks, 128, lds64, stream>>>(
            w.f, HID, W1 + (size_t)i * HID * HID, b1 + i * HID,
            w.z, HID, 0, N_NODES, HID);
        edge_softmax_aggregate(w, src, dst, a1 + i * 2 * HID, HID, HID,
                               w.h, 256, i * HID, 1.0f, stream);
    }

    // ---------------- f2 = gcn_attn(h1[N,256], W1h[256,64])  (reuses w.f)
    // GEMM first (reads w.h only), then zero w.f, then aggregate into it.
    gemm_wmma_f32_d64<<<gemmBlocks, 128, lds256, stream>>>(
        w.h, 256, W1h, b1h, w.z, HID, 0, N_NODES, 4 * HID);
    fill_f32<<<(int)(((long long)N_NODES * 64 + 255) / 256), 256, 0, stream>>>(
        w.f, 0.0f, (long long)N_NODES * 64);
    edge_softmax_aggregate(w, src, dst, a1h, HID, HID, w.f, 64, 0, 1.0f, stream);

    // ---------------- out layer: 4 heads HID->7, mean via scale=0.25 scatter
    fill_f32<<<(int)(((long long)N_NODES * OUT_DIM + 255) / 256), 256, 0, stream>>>(
        w.outs, 0.0f, (long long)N_NODES * OUT_DIM);
    for (int i = 0; i < HEADS; ++i) {
        gemm_small<<<(N_NODES * OUT_DIM + 255) / 256, 256, 0, stream>>>(
            w.f, HID, Wo + (size_t)i * HID * OUT_DIM, bo + i * OUT_DIM,
            w.z, OUT_DIM, N_NODES, HID, OUT_DIM);
        edge_softmax_aggregate(w, src, dst, ao + i * 2 * OUT_DIM, OUT_DIM, OUT_DIM,
                               w.outs, OUT_DIM, 0, 0.25f, stream);
    }

    // ---------------- final linear + softmax
    final_linear_softmax<<<NB, 256, 0, stream>>>(w.outs, linW, linb, out, N_NODES);
}